// QuantumLayer_33689723470375
// MI455X (gfx1250) — compile-verified
//
#include <hip/hip_runtime.h>
#include <math.h>

#define NQ   12
#define NL   6
#define DIM  4096      // 2^12 amplitudes
#define BLK  256       // 8 wave32 per workgroup

typedef __attribute__((ext_vector_type(2))) float v2f;
typedef __attribute__((ext_vector_type(8))) float v8f;

// RX pair update: [a;b] <- [[c,-is],[-is,c]] [a;b]   (complex packed x=re,y=im)
__device__ __forceinline__ void rx_pair(v2f& a, v2f& b, float c, float s) {
    v2f na, nb;
    na.x =  c * a.x + s * b.y;
    na.y =  c * a.y - s * b.x;
    nb.x =  s * a.y + c * b.x;
    nb.y = -s * a.x + c * b.y;
    a = na; b = nb;
}

__global__ __launch_bounds__(BLK)
void qsim_kernel(const float* __restrict__ x,
                 const float* __restrict__ w,
                 float* __restrict__ out) {
    // Two interleaved-complex state buffers: 2 * 4096 * 8B = 64 KB static LDS.
    __shared__ v2f lds2[2 * DIM];
    int cur = 0;

    const int tid = threadIdx.x;
    const int b   = blockIdx.x;

    // ---------------- init: angle-encoded product state ----------------
    // amp0 = cos(h)*e^{-ih} = (c^2, -sc); amp1 = sin(h)*e^{+ih} = (sc, s^2)
    float a0r[NQ], a0i[NQ], a1r[NQ], a1i[NQ];
    #pragma unroll
    for (int q = 0; q < NQ; ++q) {
        float ang = tanhf(x[b * NQ + q]) * 3.14159265358979323846f;
        float h = 0.5f * ang;                  // |h| < pi/2: hw trans ops are exact enough
        float c = __cosf(h), s = __sinf(h);
        a0r[q] = c * c;  a0i[q] = -s * c;
        a1r[q] = s * c;  a1i[q] = s * s;
    }
    {
        v2f* st = lds2;                     // cur == 0
        #pragma unroll 1
        for (int k = 0; k < DIM / BLK; ++k) {
            int s = tid + BLK * k;
            float vr = 1.0f, vi = 0.0f;
            #pragma unroll
            for (int q = 0; q < NQ; ++q) {  // qubit q <-> bit (11-q)
                int bit = (s >> (NQ - 1 - q)) & 1;
                float ar = bit ? a1r[q] : a0r[q];
                float ai = bit ? a1i[q] : a0i[q];
                float nr = vr * ar - vi * ai;
                float ni = vr * ai + vi * ar;
                vr = nr; vi = ni;
            }
            v2f amp; amp.x = vr; amp.y = vi;
            st[s] = amp;
        }
    }

    // ---------------- 6 entangler layers ----------------
    #pragma unroll 1
    for (int l = 0; l < NL; ++l) {
        // 12 RX gates, fused two-qubits-per-pass (RX gates commute):
        // 6 full-state passes, each thread owns disjoint amplitude quads.
        #pragma unroll 1
        for (int g = 0; g < NQ / 2; ++g) {
            __syncthreads();
            float ha = 0.5f * w[l * NQ + 2 * g];
            float hb = 0.5f * w[l * NQ + 2 * g + 1];
            float ca = __cosf(ha), sa = __sinf(ha);   // |angle| ~ 0.01 rad
            float cb = __cosf(hb), sb = __sinf(hb);
            int ma = 1 << (NQ - 1 - 2 * g);     // qubit 2g   (higher bit)
            int mb = ma >> 1;                   // qubit 2g+1 (adjacent lower bit)
            v2f* st = lds2 + cur * DIM;
            #pragma unroll
            for (int k = 0; k < (DIM / 4) / BLK; ++k) {
                int p   = tid + BLK * k;        // quad index, 2 zero bits inserted
                int s00 = ((p & ~(mb - 1)) << 2) | (p & (mb - 1));
                v2f a00 = st[s00];
                v2f a01 = st[s00 | mb];
                v2f a10 = st[s00 | ma];
                v2f a11 = st[s00 | ma | mb];
                rx_pair(a00, a10, ca, sa);      // RX on qubit 2g
                rx_pair(a01, a11, ca, sa);
                rx_pair(a00, a01, cb, sb);      // RX on qubit 2g+1
                rx_pair(a10, a11, cb, sb);
                st[s00]           = a00;
                st[s00 | mb]      = a01;
                st[s00 | ma]      = a10;
                st[s00 | ma | mb] = a11;
            }
        }
        // CNOT ring fused into ONE gather pass:
        // new[s] = old[m_0(m_1(...m_11(s)))], m_i(s)=s ^ (bit_c(s)<<t)
        __syncthreads();
        {
            v2f* src = lds2 + cur * DIM;
            v2f* dst = lds2 + (cur ^ 1) * DIM;
            #pragma unroll 1
            for (int k = 0; k < DIM / BLK; ++k) {
                int s = tid + BLK * k;
                int j = s;
                #pragma unroll
                for (int i = NQ - 1; i >= 0; --i) {
                    int cpos = NQ - 1 - i;
                    int tpos = NQ - 1 - ((i + 1) % NQ);
                    j ^= ((j >> cpos) & 1) << tpos;
                }
                dst[s] = src[j];
            }
        }
        cur ^= 1;
    }

    __syncthreads();
    // ------- probabilities + sign table into the spare buffer -------
    // Spare buffer = 8192 floats: P[0..4095], R[4096..4607], Wt[4608..5631]
    float* P  = (float*)(lds2 + (cur ^ 1) * DIM);   // M[hi][lo], s = hi*64+lo
    float* R  = P + DIM;                            // 64 x 8 result tile
    float* Wt = R + 512;                            // transposed signs: Wt[c*64+lo]
    {
        v2f* st = lds2 + cur * DIM;
        #pragma unroll 1
        for (int k = 0; k < DIM / BLK; ++k) {
            int s = tid + BLK * k;
            v2f a = st[s];
            P[s] = a.x * a.x + a.y * a.y;
        }
        // Wt[c][lo]: c<6 -> sign of bit c of lo; c==6 -> 1; else 0
        #pragma unroll
        for (int e = 0; e < 4; ++e) {
            int idx = tid + BLK * e;        // 1024 entries
            int c   = idx >> 6;
            int lo  = idx & 63;
            float v = 0.0f;
            if (c < 6)       v = 1.0f - 2.0f * (float)((lo >> c) & 1);
            else if (c == 6) v = 1.0f;
            Wt[idx] = v;
        }
    }
    __syncthreads();

    // ---------------- WMMA measurement GEMM (exact, f32) ----------------
    // R[hi][c] = sum_lo P[hi*64+lo] * Wt[c*64+lo]  via V_WMMA_F32_16X16X4_F32.
    // Waves 0..3 each own one 16-row tile; branchless b64 fragment loads.
    int wave = tid >> 5;
    int lane = tid & 31;
    if (wave < 4) {
        int row   = (wave << 4) + (lane & 15);  // A: M rows (hi)
        int col   = lane & 15;                  // B/D: N columns
        int khalf = (lane >> 4) << 1;           // lanes 0-15: K+0/1, 16-31: K+2/3
        v8f acc = {};
        #pragma unroll
        for (int k = 0; k < 64; k += 4) {
            int k0 = k + khalf;
            v2f A = *(const v2f*)(P  + row * 64 + k0);
            v2f B = *(const v2f*)(Wt + col * 64 + k0);
            acc = __builtin_amdgcn_wmma_f32_16x16x4_f32(
                /*neg_a=*/false, A, /*neg_b=*/false, B,
                /*c_mod=*/(short)0, acc, /*reuse_a=*/false, /*reuse_b=*/false);
        }
        // D layout: VGPR v holds row v (lanes 0-15) / v+8 (lanes 16-31)
        if (col < 8) {
            int rbase = (wave << 4) + ((lane >> 4) << 3);
            #pragma unroll
            for (int v = 0; v < 8; ++v)
                R[(rbase + v) * 8 + col] = acc[v];
        }
    }
    __syncthreads();

    // ---------------- fold R into 12 expectation values ----------------
    if (tid < NQ) {
        int q = tid;
        float acc = 0.0f;
        if (q >= 6) {                   // qubit lives in 'lo' bits
            int c = NQ - 1 - q;         // 0..5
            #pragma unroll 1
            for (int hi = 0; hi < 64; ++hi) acc += R[hi * 8 + c];
        } else {                        // qubit lives in 'hi' bits
            int bp = 5 - q;
            #pragma unroll 1
            for (int hi = 0; hi < 64; ++hi) {
                float sgn = 1.0f - 2.0f * (float)((hi >> bp) & 1);
                acc += sgn * R[hi * 8 + 6];     // ones column
            }
        }
        out[b * NQ + q] = acc;
    }
}

extern "C" void kernel_launch(void* const* d_in, const int* in_sizes, int n_in,
                              void* d_out, int out_size, void* d_ws, size_t ws_size,
                              hipStream_t stream) {
    const float* x  = (const float*)d_in[0];   // (2048, 12) f32
    const float* w  = (const float*)d_in[1];   // (6, 12) f32
    float* out = (float*)d_out;                // (2048, 12) f32
    int batch = in_sizes[0] / NQ;
    qsim_kernel<<<batch, BLK, 0, stream>>>(x, w, out);
}